// NeuSVolumeRendering_30468497998288
// MI455X (gfx1250) — compile-verified
//
#include <hip/hip_runtime.h>
#include <hip/hip_bf16.h>
#include <stdint.h>

// NeuS volume rendering weights: alpha from SDF/sigmoid + per-ray cumprod.
// Memory-bound (302 MB @ 23.3 TB/s ~= 13 us); design: one wave32 per ray,
// async global->LDS (CDNA5 GLOBAL_LOAD_ASYNC_TO_LDS_B128 / ASYNCcnt) staging,
// wave-level multiplicative scan for the transmittance cumprod.

#define RAYS_PER_BLOCK 8       // 8 waves of 32 = 256 threads
#define SMEM_FLOATS_PER_RAY 1024  // dir 384 | grad 384 | sdf 128 | delta 128
// per-ray LDS byte layout: dir @0 (1536B), grad @1536 (1536B), sdf @3072 (512B), delta @3584 (512B)

__device__ __forceinline__ void async_ld_b128(uint32_t lds_byte_addr,
                                              uint32_t glb_byte_off,
                                              const void* sbase) {
  // GVS mode: mem_addr = SGPR64 + VGPR32 ; vdst VGPR holds LDS byte address.
  asm volatile("global_load_async_to_lds_b128 %0, %1, %2"
               :: "v"(lds_byte_addr), "v"(glb_byte_off), "s"(sbase)
               : "memory");
}

__device__ __forceinline__ float alpha_of(float tc, float sdfv, float dt, float s) {
  // iter_cos with COS_ANNEAL_RATIO = 0.5 baked in:
  // -(relu(0.5 - 0.5*tc)*0.5 + relu(-tc)*0.5)
  float ic = -(0.5f * fmaxf(0.5f - 0.5f * tc, 0.0f) + 0.5f * fmaxf(-tc, 0.0f));
  float hs = ic * dt * 0.5f;
  float prev = 1.0f / (1.0f + expf(-((sdfv - hs) * s)));
  float next = 1.0f / (1.0f + expf(-((sdfv + hs) * s)));
  float a = (prev - next + 1e-5f) / (prev + 1e-5f);
  return fminf(fmaxf(a, 0.0f), 1.0f);
}

__global__ __launch_bounds__(RAYS_PER_BLOCK * 32)
void neus_weights_kernel(const float* __restrict__ dirs,
                         const float* __restrict__ grads,
                         const float* __restrict__ sdf,
                         const float* __restrict__ deltas,
                         const float* __restrict__ variance,
                         float* __restrict__ out,
                         int n_rays) {
  __shared__ __align__(16) float smem[RAYS_PER_BLOCK * SMEM_FLOATS_PER_RAY]; // 32 KB

  const int lane = threadIdx.x & 31;
  const int wave = threadIdx.x >> 5;
  const int ray  = blockIdx.x * RAYS_PER_BLOCK + wave;
  if (ray >= n_rays) return;

  // --- Stage this ray's data into this wave's private LDS slice (async DMA) ---
  const uint32_t slice = (uint32_t)(uintptr_t)(&smem[0]) + (uint32_t)wave * 4096u;
  const uint32_t l16   = (uint32_t)lane * 16u;
  const uint32_t goff3 = (uint32_t)ray * 1536u; // dir/grad: 128*3*4 bytes per ray
  const uint32_t goff1 = (uint32_t)ray * 512u;  // sdf/delta: 128*4 bytes per ray

#pragma unroll
  for (uint32_t c = 0; c < 3; ++c) {
    async_ld_b128(slice +          c * 512u + l16, goff3 + c * 512u + l16, dirs);
    async_ld_b128(slice + 1536u +  c * 512u + l16, goff3 + c * 512u + l16, grads);
  }
  async_ld_b128(slice + 3072u + l16, goff1 + l16, sdf);
  async_ld_b128(slice + 3584u + l16, goff1 + l16, deltas);

  asm volatile("s_wait_asynccnt 0x0" ::: "memory"); // wave-private counter: our 8 DMAs done

  // --- Read back from LDS: lane covers samples 4*lane .. 4*lane+3 ---
  const float* base = smem + wave * SMEM_FLOATS_PER_RAY;
  const float4* dp = reinterpret_cast<const float4*>(base +        lane * 12);
  const float4* gp = reinterpret_cast<const float4*>(base + 384 +  lane * 12);
  float4 d0 = dp[0], d1 = dp[1], d2 = dp[2];
  float4 g0 = gp[0], g1 = gp[1], g2 = gp[2];
  float4 sd = *reinterpret_cast<const float4*>(base + 768 + lane * 4);
  float4 dl = *reinterpret_cast<const float4*>(base + 896 + lane * 4);

  const float s = expf(variance[0] * 10.0f);

  // 3-wide dot products (xyz interleaved across the three float4s)
  float tc0 = d0.x * g0.x + d0.y * g0.y + d0.z * g0.z;
  float tc1 = d0.w * g0.w + d1.x * g1.x + d1.y * g1.y;
  float tc2 = d1.z * g1.z + d1.w * g1.w + d2.x * g2.x;
  float tc3 = d2.y * g2.y + d2.z * g2.z + d2.w * g2.w;

  float a0 = alpha_of(tc0, sd.x, dl.x, s);
  float a1 = alpha_of(tc1, sd.y, dl.y, s);
  float a2 = alpha_of(tc2, sd.z, dl.z, s);
  float a3 = alpha_of(tc3, sd.w, dl.w, s);

  float f0 = 1.0f - a0 + 1e-7f;
  float f1 = 1.0f - a1 + 1e-7f;
  float f2 = 1.0f - a2 + 1e-7f;
  float f3 = 1.0f - a3 + 1e-7f;

  // --- Wave-wide multiplicative exclusive scan of per-lane products ---
  float p = f0 * f1 * f2 * f3;
  float scan = p;
#pragma unroll
  for (int off = 1; off < 32; off <<= 1) {
    float up = __shfl_up(scan, (unsigned)off, 32);
    if (lane >= off) scan *= up;
  }
  float T = __shfl_up(scan, 1u, 32);
  if (lane == 0) T = 1.0f;

  float4 w;
  w.x = a0 * T; T *= f0;
  w.y = a1 * T; T *= f1;
  w.z = a2 * T; T *= f2;
  w.w = a3 * T;

  *reinterpret_cast<float4*>(out + (size_t)ray * 128 + lane * 4) = w;
}

extern "C" void kernel_launch(void* const* d_in, const int* in_sizes, int n_in,
                              void* d_out, int out_size, void* d_ws, size_t ws_size,
                              hipStream_t stream) {
  (void)n_in; (void)d_ws; (void)ws_size; (void)out_size;
  const float* dirs     = (const float*)d_in[0];
  const float* grads    = (const float*)d_in[1];
  const float* sdf      = (const float*)d_in[2];
  const float* deltas   = (const float*)d_in[3];
  const float* variance = (const float*)d_in[4];
  float* out = (float*)d_out;

  const int n_rays = in_sizes[2] / 128;  // sdf is [N, 128, 1]
  const int blocks = (n_rays + RAYS_PER_BLOCK - 1) / RAYS_PER_BLOCK;
  neus_weights_kernel<<<blocks, RAYS_PER_BLOCK * 32, 0, stream>>>(
      dirs, grads, sdf, deltas, variance, out, n_rays);
}